// GCNClient_48936857370856
// MI455X (gfx1250) — compile-verified
//
#include <hip/hip_runtime.h>

typedef __attribute__((ext_vector_type(2))) float v2f;
typedef __attribute__((ext_vector_type(8))) float v8f;

#define GCN_N 50000
#define GCN_E 800000
#define GCN_D 128

// ---------------- degree / normalization ----------------

__global__ void gcn_deg_init(float* __restrict__ deg, int n) {
  int i = blockIdx.x * blockDim.x + threadIdx.x;
  if (i < n) deg[i] = 1.0f;  // self-loop contribution
}

__global__ void gcn_deg_accum(const int* __restrict__ dst, float* __restrict__ deg, int nE) {
  int e = blockIdx.x * blockDim.x + threadIdx.x;
  if (e < nE) atomicAdd(&deg[dst[e]], 1.0f);
}

__global__ void gcn_dinv(const float* __restrict__ deg, float* __restrict__ dinv, int n) {
  int i = blockIdx.x * blockDim.x + threadIdx.x;
  if (i < n) dinv[i] = rsqrtf(fmaxf(deg[i], 1.0f));
}

// ---------------- H = X @ W via V_WMMA_F32_16X16X4_F32 ----------------
// Block = 256 threads = 8 waves. Block b computes rows [16b,16b+16);
// wave w computes cols [16w,16w+16). K=128 accumulated in steps of 4.
// A (16x4 f32) lane layout: lane L holds row m0+(L%16), K = k+2*(L/16)+{0,1} (float2).
// B (4x16 f32) lane layout: lane L holds col n0+(L%16), K = k+2*(L/16)+{0,1}.
// C/D (16x16 f32): VGPR r -> row m0 + r + 8*(L/16), col n0 + (L%16).

__global__ void gcn_gemm_wmma(const float* __restrict__ X, const float* __restrict__ W,
                              float* __restrict__ H) {
  const int wave = threadIdx.x >> 5;
  const int lane = threadIdx.x & 31;
  const int lo = lane & 15;
  const int hi = lane >> 4;  // 0 or 1
  const int m0 = blockIdx.x * 16;
  const int n0 = wave * 16;

  const float* arow = X + (size_t)(m0 + lo) * GCN_D + 2 * hi;
  const float* bcol = W + (size_t)(2 * hi) * GCN_D + (n0 + lo);

  v8f acc = {};
#pragma unroll
  for (int k = 0; k < GCN_D; k += 4) {
    v2f a = *(const v2f*)(arow + k);
    v2f b;
    b.x = bcol[(size_t)k * GCN_D];
    b.y = bcol[(size_t)k * GCN_D + GCN_D];
    acc = __builtin_amdgcn_wmma_f32_16x16x4_f32(
        /*neg_a=*/false, a, /*neg_b=*/false, b,
        /*c_mod=*/(short)0, acc, /*reuse_a=*/false, /*reuse_b=*/false);
  }

  float* orow = H + (size_t)(m0 + 8 * hi) * GCN_D + (n0 + lo);
#pragma unroll
  for (int r = 0; r < 8; ++r) orow[(size_t)r * GCN_D] = acc[r];
}

// ---------------- P[i] = dinv[i]^2 * H[i]  (self-loop term, full init) -------

__global__ void gcn_selfloop_init(const float* __restrict__ H, const float* __restrict__ dinv,
                                  float* __restrict__ P) {
  int idx = blockIdx.x * blockDim.x + threadIdx.x;  // over N*32 float4s
  if (idx < GCN_N * (GCN_D / 4)) {
    int node = idx >> 5;
    float s = dinv[node];
    s = s * s;
    float4 v = ((const float4*)H)[idx];
    float4 o;
    o.x = v.x * s; o.y = v.y * s; o.z = v.z * s; o.w = v.w * s;
    ((float4*)P)[idx] = o;
  }
}

// ---------------- scatter: P[dst] += dinv[src]*dinv[dst] * H[src] ------------
// One wave32 per edge; lane l owns float4 #l of the 128-dim feature:
// 512B coalesced gather + 512B coalesced f32 atomics per edge.

__global__ void gcn_scatter(const float* __restrict__ H, const int* __restrict__ src,
                            const int* __restrict__ dst, const float* __restrict__ dinv,
                            float* __restrict__ P, int nE) {
  const int wave = threadIdx.x >> 5;
  const int lane = threadIdx.x & 31;
  const int e = blockIdx.x * 8 + wave;
  if (e >= nE) return;
  const int s = src[e];
  const int d = dst[e];
  const float norm = dinv[s] * dinv[d];
  float4 v = ((const float4*)(H + (size_t)s * GCN_D))[lane];
  float* op = P + (size_t)d * GCN_D + lane * 4;
  atomicAdd(op + 0, v.x * norm);
  atomicAdd(op + 1, v.y * norm);
  atomicAdd(op + 2, v.z * norm);
  atomicAdd(op + 3, v.w * norm);
}

// ---------------- out = relu(P + b) ----------------

__global__ void gcn_bias_relu(const float* __restrict__ P, const float* __restrict__ b,
                              float* __restrict__ out) {
  int idx = blockIdx.x * blockDim.x + threadIdx.x;  // over N*32 float4s
  if (idx < GCN_N * (GCN_D / 4)) {
    float4 v = ((const float4*)P)[idx];
    float4 bb = ((const float4*)b)[idx & 31];
    float4 o;
    o.x = fmaxf(v.x + bb.x, 0.0f);
    o.y = fmaxf(v.y + bb.y, 0.0f);
    o.z = fmaxf(v.z + bb.z, 0.0f);
    o.w = fmaxf(v.w + bb.w, 0.0f);
    ((float4*)out)[idx] = o;
  }
}

// ---------------- driver ----------------

extern "C" void kernel_launch(void* const* d_in, const int* in_sizes, int n_in,
                              void* d_out, int out_size, void* d_ws, size_t ws_size,
                              hipStream_t stream) {
  const float* x  = (const float*)d_in[0];            // [N, D]
  const int*   ei = (const int*)d_in[1];              // [2, E]
  const float* W1 = (const float*)d_in[2];            // [D, D]
  const float* b1 = (const float*)d_in[3];            // [D]
  const float* W2 = (const float*)d_in[4];            // [D, D]
  const float* b2 = (const float*)d_in[5];            // [D]
  float* out = (float*)d_out;                         // [N, D]

  const int* src = ei;
  const int* dst = ei + GCN_E;

  // workspace layout (floats), 16B-aligned chunks
  float* deg  = (float*)d_ws;                 // N      (padded to 50048)
  float* dinv = deg + 50048;                  // N      (padded to 50048)
  float* bufA = dinv + 50048;                 // N*D
  float* bufB = bufA + (size_t)GCN_N * GCN_D; // N*D

  const int TPB = 256;
  const int gN     = (GCN_N + TPB - 1) / TPB;
  const int gE     = (GCN_E + TPB - 1) / TPB;
  const int gFeat  = (GCN_N * (GCN_D / 4) + TPB - 1) / TPB;  // N*32 float4s
  const int gGemm  = GCN_N / 16;                              // 3125 exact
  const int gScat  = (GCN_E + 7) / 8;                         // 8 edges/block

  // normalization: deg (with self-loops) -> dinv
  gcn_deg_init<<<gN, TPB, 0, stream>>>(deg, GCN_N);
  gcn_deg_accum<<<gE, TPB, 0, stream>>>(dst, deg, GCN_E);
  gcn_dinv<<<gN, TPB, 0, stream>>>(deg, dinv, GCN_N);

  // ---- layer 1 ----
  gcn_gemm_wmma<<<gGemm, TPB, 0, stream>>>(x, W1, bufA);            // H1 = x @ W1
  gcn_selfloop_init<<<gFeat, TPB, 0, stream>>>(bufA, dinv, bufB);   // P1 = dinv^2 * H1
  gcn_scatter<<<gScat, TPB, 0, stream>>>(bufA, src, dst, dinv, bufB, GCN_E);
  gcn_bias_relu<<<gFeat, TPB, 0, stream>>>(bufB, b1, bufA);         // A1 = relu(P1 + b1)

  // ---- layer 2 ----
  gcn_gemm_wmma<<<gGemm, TPB, 0, stream>>>(bufA, W2, bufB);         // H2 = A1 @ W2
  gcn_selfloop_init<<<gFeat, TPB, 0, stream>>>(bufB, dinv, bufA);   // P2 = dinv^2 * H2
  gcn_scatter<<<gScat, TPB, 0, stream>>>(bufB, src, dst, dinv, bufA, GCN_E);
  gcn_bias_relu<<<gFeat, TPB, 0, stream>>>(bufA, b2, out);          // out = relu(P2 + b2)
}